// Temporal_Attention_35983236006189
// MI455X (gfx1250) — compile-verified
//
#include <hip/hip_runtime.h>
#include <math.h>

typedef __attribute__((ext_vector_type(2))) float v2f;
typedef __attribute__((ext_vector_type(8))) float v8f;

static constexpr int nB  = 64;
static constexpr int nT  = 256;
static constexpr int nD  = 128;
static constexpr int nH  = 8;
static constexpr int nE  = 128;
static constexpr int nHE = 1024;

// One wave computes a 16x16 fp32 tile: acc += A(16xK) * B(Kx16)
// A[m,k] = Ap[m*lda + k]          (k contiguous)
// B[k,n] = Bp[k*bks + n*bns]
// Layouts (wave32):
//   A frag (v2f): lane m = lane&15, K = 2*(lane>>4) + vgpr
//   B frag (v2f): lane n = lane&15, K = 2*(lane>>4) + vgpr
//   C/D  (v8f):  lane n = lane&15, M = r + 8*(lane>>4)
__device__ __forceinline__ v8f wmma_tile_f32(const float* __restrict__ Ap, int lda,
                                             const float* __restrict__ Bp, int bks, int bns,
                                             int K, v8f acc, int lane) {
  const int half = lane >> 4;
  const int mn   = lane & 15;
  const float* arow = Ap + mn * lda;
  const float* bcol = Bp + mn * bns;
  for (int k0 = 0; k0 < K; k0 += 4) {
    const int ka = k0 + 2 * half;
    v2f a, b;
    a.x = arow[ka];
    a.y = arow[ka + 1];
    b.x = bcol[ka * bks];
    b.y = bcol[(ka + 1) * bks];
    acc = __builtin_amdgcn_wmma_f32_16x16x4_f32(false, a, false, b,
                                                (short)0, acc, false, false);
  }
  return acc;
}

__device__ __forceinline__ void store_tile_f32(float* __restrict__ Cp, int ldc,
                                               v8f acc, int lane) {
  const int half = lane >> 4;
  const int n    = lane & 15;
#pragma unroll
  for (int r = 0; r < 8; ++r)
    Cp[(r + 8 * half) * ldc + n] = acc[r];
}

// ---- K1: xs[b,d] = sum_t Ws[t]*x[b,t,d];  S = sum_t Ws[t] ----
__global__ void k_reduce_x(const float* __restrict__ x, const float* __restrict__ Ws,
                           float* __restrict__ xs, float* __restrict__ Sp) {
  const int b = blockIdx.x;
  const int d = threadIdx.x;                       // 128 threads
  const float* xb = x + (size_t)b * nT * nD + d;
  float acc = 0.f;
  for (int t = 0; t < nT; ++t) acc = fmaf(Ws[t], xb[t * nD], acc);
  xs[b * nD + d] = acc;
  if (b == 0 && d == 0) {
    float s = 0.f;
    for (int t = 0; t < nT; ++t) s += Ws[t];
    *Sp = s;
  }
}

// ---- K2: Ksum[64,1024] = xs[64,128] @ Wk[128,1024] + S*bk ----
__global__ void k_ksum(const float* __restrict__ xs, const float* __restrict__ Wk,
                       const float* __restrict__ bk, const float* __restrict__ Sp,
                       float* __restrict__ Ksum) {
  const int lane = threadIdx.x & 31;
  const int tile = blockIdx.x * (blockDim.x >> 5) + (threadIdx.x >> 5); // 0..255
  const int m0 = (tile >> 6) * 16;                 // 4 M-tiles
  const int n0 = (tile & 63) * 16;                 // 64 N-tiles
  const float S = *Sp;
  const float bias = S * bk[n0 + (lane & 15)];
  v8f acc;
#pragma unroll
  for (int r = 0; r < 8; ++r) acc[r] = bias;
  acc = wmma_tile_f32(xs + m0 * nD, nD, Wk + n0, nHE, 1, nD, acc, lane);
  store_tile_f32(Ksum + m0 * nHE + n0, nHE, acc, lane);
}

// ---- K3: wq_eff[b,h,d] = sum_e Ksum[b,h,e] * Wq[d, h*E+e]  (B transposed access) ----
__global__ void k_wqeff(const float* __restrict__ Ksum, const float* __restrict__ Wq,
                        float* __restrict__ wq) {
  const int lane = threadIdx.x & 31;
  const int tile = blockIdx.x * (blockDim.x >> 5) + (threadIdx.x >> 5); // 0..255
  const int h  = tile >> 5;
  const int r  = tile & 31;
  const int m0 = (r >> 3) * 16;                    // 4 M-tiles (b)
  const int n0 = (r & 7) * 16;                     // 8 N-tiles (d)
  v8f acc = {};
  // A[m,k] = Ksum[(m0+m)*HE + h*E + k] ; B[k,n] = Wq[(n0+n)*HE + h*E + k]
  acc = wmma_tile_f32(Ksum + m0 * nHE + h * nE, nHE,
                      Wq + n0 * nHE + h * nE, 1, nHE, nE, acc, lane);
  store_tile_f32(wq + m0 * nHE + h * nE + n0, nHE, acc, lane);
}

// ---- K4: per-batch summary -> softmax over t -> xv[b,h,d] = sum_t beta*x ----
__global__ void k_attn(const float* __restrict__ x, const float* __restrict__ wq,
                       float* __restrict__ xv) {
  const int b = blockIdx.x;
  const int tid = threadIdx.x;                     // 256 threads = 8 waves
  __shared__ float s_wq[nH * nD];                  // 4 KB
  __shared__ float s_s[nH][nT];                    // 8 KB
  for (int i = tid; i < nH * nD; i += 256) s_wq[i] = wq[b * nHE + i];
  __syncthreads();

  { // phase 1: s_s[h][t] = scale * x[b,t,:] . wq_eff[b,h,:]
    const int t = tid;
    const float* xt = x + ((size_t)b * nT + t) * nD;
    float acc[nH];
#pragma unroll
    for (int h = 0; h < nH; ++h) acc[h] = 0.f;
    for (int d = 0; d < nD; ++d) {
      const float xd = xt[d];
#pragma unroll
      for (int h = 0; h < nH; ++h) acc[h] = fmaf(xd, s_wq[h * nD + d], acc[h]);
    }
    const float scale = 0.08838834764831845f;      // 1/sqrt(128)
#pragma unroll
    for (int h = 0; h < nH; ++h) s_s[h][t] = scale * acc[h];
  }
  __syncthreads();

  { // phase 2: softmax over t; wave w handles head h=w (wave32)
    const int w = tid >> 5, lane = tid & 31;
    float m = -3.402823466e38f;
    for (int i = lane; i < nT; i += 32) m = fmaxf(m, s_s[w][i]);
#pragma unroll
    for (int o = 16; o > 0; o >>= 1) m = fmaxf(m, __shfl_xor(m, o, 32));
    float sum = 0.f;
    for (int i = lane; i < nT; i += 32) {
      const float e = __expf(s_s[w][i] - m);
      s_s[w][i] = e;
      sum += e;
    }
#pragma unroll
    for (int o = 16; o > 0; o >>= 1) sum += __shfl_xor(sum, o, 32);
    const float inv = 1.f / sum;
    for (int i = lane; i < nT; i += 32) s_s[w][i] *= inv;
  }
  __syncthreads();

  // phase 3: xv[b,h,d] = sum_t beta[h][t] * x[b,t,d]
  for (int idx = tid; idx < nHE; idx += 256) {
    const int h = idx >> 7, d = idx & 127;
    const float* xb = x + (size_t)b * nT * nD + d;
    float acc = 0.f;
    for (int t = 0; t < nT; ++t) acc = fmaf(s_s[h][t], xb[t * nD], acc);
    xv[b * nHE + idx] = acc;
  }
}

// ---- K5: result[b, h*E+e] = xv[b,h,:] @ Wv[:, h*E+e] + bv ----
__global__ void k_vproj(const float* __restrict__ xv, const float* __restrict__ Wv,
                        const float* __restrict__ bv, float* __restrict__ result) {
  const int lane = threadIdx.x & 31;
  const int tile = blockIdx.x * (blockDim.x >> 5) + (threadIdx.x >> 5); // 0..255
  const int h  = tile >> 5;
  const int r  = tile & 31;
  const int m0 = (r >> 3) * 16;
  const int n0 = (r & 7) * 16;
  const float bias = bv[h * nE + n0 + (lane & 15)];
  v8f acc;
#pragma unroll
  for (int rr = 0; rr < 8; ++rr) acc[rr] = bias;
  // A[m,k] = xv[(m0+m)*HE + h*E + k] ; B[k,n] = Wv[k*HE + h*E + n0+n]
  acc = wmma_tile_f32(xv + m0 * nHE + h * nE, nHE,
                      Wv + h * nE + n0, nHE, 1, nD, acc, lane);
  store_tile_f32(result + m0 * nHE + h * nE + n0, nHE, acc, lane);
}

// ---- K6: out[64,128] = result[64,1024] @ Wo[1024,128] + bo ----
__global__ void k_oproj(const float* __restrict__ result, const float* __restrict__ Wo,
                        const float* __restrict__ bo, float* __restrict__ out) {
  const int lane = threadIdx.x & 31;
  const int tile = blockIdx.x * (blockDim.x >> 5) + (threadIdx.x >> 5); // 0..31
  const int m0 = (tile >> 3) * 16;
  const int n0 = (tile & 7) * 16;
  const float bias = bo[n0 + (lane & 15)];
  v8f acc;
#pragma unroll
  for (int rr = 0; rr < 8; ++rr) acc[rr] = bias;
  acc = wmma_tile_f32(result + m0 * nHE, nHE, Wo + n0, nD, 1, nHE, acc, lane);
  store_tile_f32(out + m0 * nD + n0, nD, acc, lane);
}

extern "C" void kernel_launch(void* const* d_in, const int* in_sizes, int n_in,
                              void* d_out, int out_size, void* d_ws, size_t ws_size,
                              hipStream_t stream) {
  const float* x  = (const float*)d_in[0];
  const float* Wq = (const float*)d_in[1];
  // d_in[2] = bq : cancels in softmax (constant over t)
  const float* Wk = (const float*)d_in[3];
  const float* bk = (const float*)d_in[4];
  const float* Wv = (const float*)d_in[5];
  const float* bv = (const float*)d_in[6];
  const float* Ws = (const float*)d_in[7];
  // d_in[8] = bs : cancels in softmax (constant over t)
  const float* Wo = (const float*)d_in[9];
  const float* bo = (const float*)d_in[10];
  float* out = (float*)d_out;

  float* ws     = (float*)d_ws;
  float* xs     = ws;                 // 8192
  float* Sp     = ws + 8192;          // 1 (padded region to 8448)
  float* Ksum   = ws + 8448;          // 65536
  float* wqeff  = Ksum + 65536;       // 65536
  float* xv     = wqeff + 65536;      // 65536
  float* result = xv + 65536;         // 65536  (total ~1.03 MB)

  k_reduce_x<<<nB, nD, 0, stream>>>(x, Ws, xs, Sp);
  k_ksum    <<<32, 256, 0, stream>>>(xs, Wk, bk, Sp, Ksum);
  k_wqeff   <<<32, 256, 0, stream>>>(Ksum, Wq, wqeff);
  k_attn    <<<nB, 256, 0, stream>>>(x, wqeff, xv);
  k_vproj   <<<32, 256, 0, stream>>>(xv, Wv, bv, result);
  k_oproj   <<<4, 256, 0, stream>>>(result, Wo, bo, out);
}